// TTT_2929167696270
// MI455X (gfx1250) — compile-verified
//
#include <hip/hip_runtime.h>
#include <hip/hip_bf16.h>
#include <math.h>

// CDNA5 wave32 fp32 WMMA: D(16x16,f32) = A(16x4,f32) x B(4x16,f32) + C
typedef float v2f __attribute__((ext_vector_type(2)));
typedef float v8f __attribute__((ext_vector_type(8)));
typedef unsigned int v4u __attribute__((ext_vector_type(4)));
typedef int v4i __attribute__((ext_vector_type(4)));
typedef int v8i __attribute__((ext_vector_type(8)));

#define WMMA4(A_, B_, C_) \
  __builtin_amdgcn_wmma_f32_16x16x4_f32(false, (A_), false, (B_), (short)0, (C_), false, false)

// Problem constants (fixed by the reference)
#define BB   32
#define NN   1024
#define CC   768
#define HH   12
#define DD   64
#define QKVN 2496          // 3C + 3D
#define CONC 832           // C + D
#define ESC  (-(1.0f/3.0f) / 1024.0f)   // -SCALE / N,  SCALE = 9^-0.5 = 1/3

// ---------------------------------------------------------------------------
// Tensor Data Mover: async copy of a 2-D fp32 tile (tileX x tileY, row stride
// `strideElems` in elements) from global memory into contiguous LDS at byte
// offset `ldsOff`.  D# built per CDNA5 ISA 8.3/8.4:
//   group0: count=1 | lds_addr | global_addr[56:0] | type=2
//   group1: data_size=4B, tensor_dim0=tile_dim0=tileX, tensor_dim1=tile_dim1=
//           tileY (dims==tile dims -> no OOB clipping), dim0_stride=strideElems
//   group2/3 + trailing group: zero (<=2D tensor, no gather/iterate)
// Issue from ONE wave only; pair with s_wait_tensorcnt(0) in that wave, then a
// workgroup barrier. (EXEC is ignored by tensor instructions.)
// This toolchain exposes the 6-arg builtin:
//   (uint32x4 g0, int32x8 g1, int32x4 g2, int32x4 g3, int32x8 gx, i32 cpol)
// ---------------------------------------------------------------------------
static __device__ __forceinline__ void tdm_load_tile_f32(uint32_t ldsOff,
                                                         const float* gsrc,
                                                         uint32_t tileX,
                                                         uint32_t tileY,
                                                         uint32_t strideElems)
{
    const uint64_t ga = (uint64_t)(uintptr_t)gsrc;
    v4u g0;
    g0[0] = 1u;                                    // count=1, user mode, no gather
    g0[1] = ldsOff;                                // LDS byte address
    g0[2] = (uint32_t)ga;                          // global_addr[31:0]
    g0[3] = ((uint32_t)(ga >> 32) & 0x01FFFFFFu)   // global_addr[56:32]
            | (2u << 30);                          // type = 2
    v8i g1;
    g1[0] = (int)(2u << 16);                       // data_size=4B, no pad/multicast
    g1[1] = (int)((tileX & 0xFFFFu) << 16);        // tensor_dim0[15:0]
    g1[2] = (int)(((tileX >> 16) & 0xFFFFu) |      // tensor_dim0[31:16]
                  ((tileY & 0xFFFFu) << 16));      // tensor_dim1[15:0]
    g1[3] = (int)(((tileY >> 16) & 0xFFFFu) |      // tensor_dim1[31:16]
                  ((tileX & 0xFFFFu) << 16));      // tile_dim0
    g1[4] = (int)(tileY & 0xFFFFu);                // tile_dim1 ; tile_dim2=0
    g1[5] = (int)strideElems;                      // tensor_dim0_stride[31:0]
    g1[6] = 0;                                     // stride0[47:32] | dim1_stride lo
    g1[7] = 0;
    const v4i zero4 = {0, 0, 0, 0};
    const v8i zero8 = {0, 0, 0, 0, 0, 0, 0, 0};
    __builtin_amdgcn_tensor_load_to_lds(g0, g1, zero4, zero4, zero8, 0);
}

// ---------------------------------------------------------------------------
// Generic WMMA GEMM:  C[M,N] = A[M,K] @ B[K,N] + bias[N]
// Requires: M % 128 == 0, N % 64 == 0, K % 16 == 0.
// Block = 256 threads (8 waves), block tile 128x64, wave tile 32x32.
// B tile (16x64, contiguous) is staged by the TDM; A tile by vector loads into
// a bank-spread (stride 20) LDS layout.  WMMA fragment layouts per ISA 7.12.2:
//   A 16x4 : lanes 0-15 -> M=lane, K=kb(+1); lanes 16-31 -> K=kb+2(+3)
//   B 4x16 : vgpr0 lanes0-15 K=kb, lanes16-31 K=kb+2 ; vgpr1 K=kb+1 / kb+3
//   C 16x16: vgpr r -> M = r (lanes0-15) / r+8 (lanes16-31), N = lane&15
// ---------------------------------------------------------------------------
__global__ __launch_bounds__(256)
void gemm_bias_f32(const float* __restrict__ A, const float* __restrict__ Bm,
                   const float* __restrict__ bias, float* __restrict__ Cm,
                   int M, int N, int K)
{
    __shared__ __align__(16) float As[128 * 20];   // 128 rows x 16 k, stride 20
    __shared__ __align__(16) float Bs[16 * 64];    // 16 k x 64 cols (contiguous)

    const int t     = threadIdx.x;
    const int lane  = t & 31;
    const int l15   = lane & 15;
    const int lhalf = lane >> 4;
    const int wave  = t >> 5;
    const int waveM = wave >> 1;   // 0..3
    const int waveN = wave & 1;    // 0..1
    const int bM    = blockIdx.y * 128;
    const int bN    = blockIdx.x * 64;
    const uint32_t bsOff = (uint32_t)(uintptr_t)(&Bs[0]);

    v8f acc[2][2] = {};

    for (int k0 = 0; k0 < K; k0 += 16) {
        // async-stage B tile 16x64 via the Tensor Data Mover (wave 0 issues)
        if (t < 32)
            tdm_load_tile_f32(bsOff, Bm + (size_t)k0 * N + bN, 64u, 16u, (uint32_t)N);

        // stage A tile 128x16 (512 float4, 2 per thread)
        #pragma unroll
        for (int i = 0; i < 2; ++i) {
            int id = t + 256 * i;
            int r  = id >> 2;
            int cg = (id & 3) * 4;
            float4 v = *(const float4*)(A + (size_t)(bM + r) * K + k0 + cg);
            *(float4*)(&As[r * 20 + cg]) = v;
        }
        // prefetch next A tile rows into cache
        if (t < 128 && k0 + 16 < K)
            __builtin_prefetch(A + (size_t)(bM + t) * K + k0 + 16, 0, 3);

        if (t < 32) __builtin_amdgcn_s_wait_tensorcnt(0);
        __syncthreads();

        #pragma unroll
        for (int kk = 0; kk < 16; kk += 4) {
            const int ka = kk + lhalf * 2;
            v2f a0, a1, b0, b1;
            a0.x = As[(waveM * 32 +      l15) * 20 + ka];
            a0.y = As[(waveM * 32 +      l15) * 20 + ka + 1];
            a1.x = As[(waveM * 32 + 16 + l15) * 20 + ka];
            a1.y = As[(waveM * 32 + 16 + l15) * 20 + ka + 1];
            const int c0 = waveN * 32 + l15;
            b0.x = Bs[ ka      * 64 + c0];
            b0.y = Bs[(ka + 1) * 64 + c0];
            b1.x = Bs[ ka      * 64 + c0 + 16];
            b1.y = Bs[(ka + 1) * 64 + c0 + 16];
            acc[0][0] = WMMA4(a0, b0, acc[0][0]);
            acc[0][1] = WMMA4(a0, b1, acc[0][1]);
            acc[1][0] = WMMA4(a1, b0, acc[1][0]);
            acc[1][1] = WMMA4(a1, b1, acc[1][1]);
        }
        __syncthreads();
    }

    #pragma unroll
    for (int mi = 0; mi < 2; ++mi) {
        #pragma unroll
        for (int ni = 0; ni < 2; ++ni) {
            const int col = bN + waveN * 32 + ni * 16 + l15;
            const float bv = bias[col];
            #pragma unroll
            for (int r = 0; r < 8; ++r) {
                const int row = bM + waveM * 32 + mi * 16 + r + lhalf * 8;
                Cm[(size_t)row * N + col] = acc[mi][ni][r] + bv;
            }
        }
    }
}

// ---------------------------------------------------------------------------
// Inner SwiGLU fast-weight update: per (b,h) block computes
//   z1 = k@w1, z2 = k@w2, t1 = e*(z2*sig), t2 = e*z1*(sig*(1+z2*(1-sig)))
//   g1 = k^T@t1, g2 = k^T@t2, column-normalize, w{1,2}u = w - g/(||g||_col+1)
// k,v read from qkv; chunked over N in 32-row chunks.
// ---------------------------------------------------------------------------
#define LDSF 2176   // 32 * 68 floats (row stride 68 for bank spread)

__global__ __launch_bounds__(256)
void ttt_swiglu_update(const float* __restrict__ qkv, const float* __restrict__ w1g,
                       const float* __restrict__ w2g, float* __restrict__ w1u,
                       float* __restrict__ w2u)
{
    __shared__ __align__(16) float lds[4 * LDSF];  // kc | z2s | t1s | t2s ; later g1s | g2s
    __shared__ float scale1[64], scale2[64];

    const int bh    = blockIdx.x;          // b*12 + h
    const int h     = bh % HH;
    const int b     = bh / HH;
    const int t     = threadIdx.x;
    const int lane  = t & 31;
    const int l15   = lane & 15;
    const int lhalf = lane >> 4;
    const int wave  = t >> 5;

    float* kc  = lds;
    float* z2s = lds + LDSF;
    float* t1s = lds + 2 * LDSF;
    float* t2s = lds + 3 * LDSF;

    const size_t rowBase = (size_t)b * NN;
    const int kcol = CC + h * DD;          // k1 columns
    const int vcol = 2 * CC + h * DD;      // v1 columns
    const float* wz = (wave < 4) ? (w1g + h * 4096) : (w2g + h * 4096);

    const int zw = wave & 3;
    const int zs = zw >> 1;                // z-phase row strip (0/1)
    const int zn = zw & 1;                 // z-phase col half (0/1)

    const int gd0 = (wave & 3) * 16;       // g-phase d strip
    const float* tsrc = (wave < 4) ? t1s : t2s;
    v8f g[4] = {};

    for (int ch = 0; ch < 32; ++ch) {
        __syncthreads();                   // prev g-phase finished with lds
        // load k chunk 32x64
        #pragma unroll
        for (int i = 0; i < 2; ++i) {
            int id = t + 256 * i;
            int r  = id >> 4;
            int cg = (id & 15) * 4;
            *(float4*)(&kc[r * 68 + cg]) =
                *(const float4*)(qkv + (rowBase + ch * 32 + r) * QKVN + kcol + cg);
        }
        __syncthreads();

        // z-phase: waves 0-3 -> z1 (kept in regs), waves 4-7 -> z2 (spilled to LDS)
        v8f z[2] = {};
        #pragma unroll
        for (int kk = 0; kk < 64; kk += 4) {
            const int ka = kk + lhalf * 2;
            v2f a, b0, b1;
            a.x = kc[(zs * 16 + l15) * 68 + ka];
            a.y = kc[(zs * 16 + l15) * 68 + ka + 1];
            const int c0 = zn * 32 + l15;
            b0.x = wz[ ka      * 64 + c0];
            b0.y = wz[(ka + 1) * 64 + c0];
            b1.x = wz[ ka      * 64 + c0 + 16];
            b1.y = wz[(ka + 1) * 64 + c0 + 16];
            z[0] = WMMA4(a, b0, z[0]);
            z[1] = WMMA4(a, b1, z[1]);
        }
        if (wave >= 4) {
            #pragma unroll
            for (int ni = 0; ni < 2; ++ni)
                #pragma unroll
                for (int r = 0; r < 8; ++r)
                    z2s[(zs * 16 + r + lhalf * 8) * 68 + zn * 32 + ni * 16 + l15] = z[ni][r];
        }
        __syncthreads();

        if (wave < 4) {                    // elementwise -> t1s, t2s
            #pragma unroll
            for (int ni = 0; ni < 2; ++ni) {
                #pragma unroll
                for (int r = 0; r < 8; ++r) {
                    const int rl  = zs * 16 + r + lhalf * 8;
                    const int cl  = zn * 32 + ni * 16 + l15;
                    const float z1v = z[ni][r];
                    const float z2v = z2s[rl * 68 + cl];
                    const float vv  = qkv[(rowBase + ch * 32 + rl) * QKVN + vcol + cl];
                    const float e   = vv * ESC;
                    const float sig = 1.0f / (1.0f + expf(-z2v));
                    t1s[rl * 68 + cl] = e * (z2v * sig);
                    t2s[rl * 68 + cl] = e * z1v * (sig * (1.0f + z2v * (1.0f - sig)));
                }
            }
        }
        __syncthreads();

        // g-phase: g(64x64) += kc^T(64x32) @ t(32x64)
        #pragma unroll
        for (int kk = 0; kk < 32; kk += 4) {
            const int ka = kk + lhalf * 2;
            v2f a;
            a.x = kc[ ka      * 68 + gd0 + l15];
            a.y = kc[(ka + 1) * 68 + gd0 + l15];
            #pragma unroll
            for (int ni = 0; ni < 4; ++ni) {
                v2f bf;
                const int c0 = ni * 16 + l15;
                bf.x = tsrc[ ka      * 68 + c0];
                bf.y = tsrc[(ka + 1) * 68 + c0];
                g[ni] = WMMA4(a, bf, g[ni]);
            }
        }
    }
    __syncthreads();

    // spill g to LDS (g1s at lds[0], g2s at lds[2*LDSF]; each 64 rows x stride 68)
    float* gs = (wave < 4) ? lds : (lds + 2 * LDSF);
    #pragma unroll
    for (int ni = 0; ni < 4; ++ni)
        #pragma unroll
        for (int r = 0; r < 8; ++r)
            gs[(gd0 + r + lhalf * 8) * 68 + ni * 16 + l15] = g[ni][r];
    __syncthreads();

    if (t < 128) {                         // per-column norms over d
        const float* gsrc = (t < 64) ? lds : (lds + 2 * LDSF);
        const int e = t & 63;
        float s = 0.f;
        for (int d2 = 0; d2 < 64; ++d2) { float v = gsrc[d2 * 68 + e]; s += v * v; }
        const float sc = 1.0f / (sqrtf(s) + 1.0f);
        if (t < 64) scale1[e] = sc; else scale2[e] = sc;
    }
    __syncthreads();

    #pragma unroll
    for (int i = 0; i < 16; ++i) {
        const int idx = t + 256 * i;       // 0..4095
        const int d2 = idx >> 6, e = idx & 63;
        w1u[(size_t)bh * 4096 + idx] = w1g[h * 4096 + idx] - lds[d2 * 68 + e] * scale1[e];
        w2u[(size_t)bh * 4096 + idx] = w2g[h * 4096 + idx] - lds[2 * LDSF + d2 * 68 + e] * scale2[e];
    }
}

// ---------------------------------------------------------------------------
// x1 = (q1 @ w1u) * silu(q1 @ w2u), written into concat[:, 0:768]
// One block per (b, h, 64-row chunk).
// ---------------------------------------------------------------------------
__global__ __launch_bounds__(256)
void ttt_x1_kernel(const float* __restrict__ qkv, const float* __restrict__ w1u,
                   const float* __restrict__ w2u, float* __restrict__ concat)
{
    __shared__ __align__(16) float qc[64 * 68];
    __shared__ __align__(16) float w1s[64 * 64];
    __shared__ __align__(16) float w2s[64 * 64];

    const int ch = blockIdx.x;             // 0..15
    const int h  = blockIdx.y;             // 0..11
    const int b  = blockIdx.z;             // 0..31
    const int bh = b * HH + h;
    const int t     = threadIdx.x;
    const int lane  = t & 31;
    const int l15   = lane & 15;
    const int lhalf = lane >> 4;
    const int wave  = t >> 5;
    const int strip = wave >> 1;           // 0..3 (16-row strips)
    const int nhalf = wave & 1;            // 0..1 (32-col halves)

    // stage w1u/w2u for this (b,h): 1024 float4 each, 4 per thread
    #pragma unroll
    for (int i = 0; i < 4; ++i) {
        const int id4 = t + 256 * i;
        *(float4*)(&w1s[id4 * 4]) = *(const float4*)(w1u + (size_t)bh * 4096 + id4 * 4);
        *(float4*)(&w2s[id4 * 4]) = *(const float4*)(w2u + (size_t)bh * 4096 + id4 * 4);
    }
    // stage q chunk 64x64
    #pragma unroll
    for (int i = 0; i < 4; ++i) {
        const int id = t + 256 * i;
        const int r  = id >> 4;
        const int cg = (id & 15) * 4;
        *(float4*)(&qc[r * 68 + cg]) =
            *(const float4*)(qkv + ((size_t)b * NN + ch * 64 + r) * QKVN + h * DD + cg);
    }
    __syncthreads();

    v8f y1[2] = {}, y2[2] = {};
    #pragma unroll
    for (int kk = 0; kk < 64; kk += 4) {
        const int ka = kk + lhalf * 2;
        v2f a;
        a.x = qc[(strip * 16 + l15) * 68 + ka];
        a.y = qc[(strip * 16 + l15) * 68 + ka + 1];
        #pragma unroll
        for (int ni = 0; ni < 2; ++ni) {
            const int c0 = nhalf * 32 + ni * 16 + l15;
            v2f bf1, bf2;
            bf1.x = w1s[ ka      * 64 + c0];
            bf1.y = w1s[(ka + 1) * 64 + c0];
            bf2.x = w2s[ ka      * 64 + c0];
            bf2.y = w2s[(ka + 1) * 64 + c0];
            y1[ni] = WMMA4(a, bf1, y1[ni]);
            y2[ni] = WMMA4(a, bf2, y2[ni]);
        }
    }

    #pragma unroll
    for (int ni = 0; ni < 2; ++ni) {
        #pragma unroll
        for (int r = 0; r < 8; ++r) {
            const size_t row = (size_t)b * NN + ch * 64 + strip * 16 + r + lhalf * 8;
            const int    col = h * DD + nhalf * 32 + ni * 16 + l15;
            const float u  = y1[ni][r];
            const float v2 = y2[ni][r];
            const float sig = 1.0f / (1.0f + expf(-v2));
            concat[row * CONC + col] = u * (v2 * sig);
        }
    }
}

// ---------------------------------------------------------------------------
// Depthwise 3x3 fast-weight update + conv: one block per (b,d) channel.
// Writes x2 into concat[:, 768:832]. Tiny workload -> plain VALU + LDS.
// ---------------------------------------------------------------------------
__global__ __launch_bounds__(256)
void ttt_dwc_kernel(const float* __restrict__ qkv, const float* __restrict__ w3g,
                    float* __restrict__ concat)
{
    __shared__ float ks[1024];
    __shared__ float es[1024];
    __shared__ float red[9 * 256];

    const int c = blockIdx.x;              // 0..2047
    const int b = c >> 6, d = c & 63;
    const int t = threadIdx.x;
    const size_t base = (size_t)b * NN;
    const int q2col = 3 * CC + d;          // 2304 + d
    const int k2col = 3 * CC + DD + d;     // 2368 + d
    const int v2col = 3 * CC + 2 * DD + d; // 2432 + d

    #pragma unroll
    for (int i = 0; i < 4; ++i) {
        const int p = t + 256 * i;
        ks[p] = qkv[(base + p) * QKVN + k2col];
        es[p] = qkv[(base + p) * QKVN + v2col] * ESC;
    }
    __syncthreads();

    float part[9];
    #pragma unroll
    for (int j = 0; j < 9; ++j) part[j] = 0.f;
    for (int i = 0; i < 4; ++i) {
        const int p = t + 256 * i;
        const int y = p >> 5, x = p & 31;
        const float ev = es[p];
        #pragma unroll
        for (int dy = -1; dy <= 1; ++dy)
            #pragma unroll
            for (int dx = -1; dx <= 1; ++dx) {
                const int yy = y + dy, xx = x + dx;
                const float kv = (yy >= 0 && yy < 32 && xx >= 0 && xx < 32)
                                   ? ks[yy * 32 + xx] : 0.f;
                part[(dy + 1) * 3 + (dx + 1)] += kv * ev;
            }
    }
    #pragma unroll
    for (int j = 0; j < 9; ++j) red[j * 256 + t] = part[j];
    __syncthreads();
    for (int s = 128; s > 0; s >>= 1) {
        if (t < s) {
            #pragma unroll
            for (int j = 0; j < 9; ++j) red[j * 256 + t] += red[j * 256 + t + s];
        }
        __syncthreads();
    }

    float wv[9];
    {
        float nrm = 0.f;
        #pragma unroll
        for (int j = 0; j < 9; ++j) { const float gv = red[j * 256]; nrm += gv * gv; }
        const float sc = 1.0f / (sqrtf(nrm) + 1.0f);
        #pragma unroll
        for (int j = 0; j < 9; ++j) wv[j] = w3g[d * 9 + j] - red[j * 256] * sc;
    }
    __syncthreads();

    #pragma unroll
    for (int i = 0; i < 4; ++i) {          // reuse ks for q2
        const int p = t + 256 * i;
        ks[p] = qkv[(base + p) * QKVN + q2col];
    }
    __syncthreads();

    for (int i = 0; i < 4; ++i) {
        const int p = t + 256 * i;
        const int y = p >> 5, x = p & 31;
        float acc = 0.f;
        #pragma unroll
        for (int dy = -1; dy <= 1; ++dy)
            #pragma unroll
            for (int dx = -1; dx <= 1; ++dx) {
                const int yy = y + dy, xx = x + dx;
                const float qv = (yy >= 0 && yy < 32 && xx >= 0 && xx < 32)
                                   ? ks[yy * 32 + xx] : 0.f;
                acc += wv[(dy + 1) * 3 + (dx + 1)] * qv;
            }
        concat[(base + p) * CONC + CC + d] = acc;
    }
}

// ---------------------------------------------------------------------------
extern "C" void kernel_launch(void* const* d_in, const int* in_sizes, int n_in,
                              void* d_out, int out_size, void* d_ws, size_t ws_size,
                              hipStream_t stream)
{
    (void)in_sizes; (void)n_in; (void)out_size; (void)ws_size;

    const float* x      = (const float*)d_in[0];   // [32,1024,768]
    const float* W_qkv  = (const float*)d_in[1];   // [768,2496]
    const float* b_qkv  = (const float*)d_in[2];   // [2496]
    const float* w1     = (const float*)d_in[3];   // [1,12,64,64]
    const float* w2     = (const float*)d_in[4];   // [1,12,64,64]
    const float* w3     = (const float*)d_in[5];   // [64,1,3,3]
    const float* proj_W = (const float*)d_in[6];   // [832,768]
    const float* proj_b = (const float*)d_in[7];   // [768]
    float* out = (float*)d_out;                    // [32,1024,768]

    // workspace layout (floats): qkv | w1u | w2u | concat  (~449 MB total)
    float* qkv    = (float*)d_ws;
    float* w1u    = qkv + (size_t)BB * NN * QKVN;          // 384*4096
    float* w2u    = w1u + (size_t)BB * HH * DD * DD;
    float* concat = w2u + (size_t)BB * HH * DD * DD;       // [32768, 832]

    // 1) qkv = x @ W_qkv + b_qkv          M=32768 N=2496 K=768
    gemm_bias_f32<<<dim3(QKVN / 64, (BB * NN) / 128), 256, 0, stream>>>(
        x, W_qkv, b_qkv, qkv, BB * NN, QKVN, CC);

    // 2) fast-weight swiglu update -> w1u, w2u
    ttt_swiglu_update<<<dim3(BB * HH), 256, 0, stream>>>(qkv, w1, w2, w1u, w2u);

    // 3) x1 -> concat[:, :768]
    ttt_x1_kernel<<<dim3(NN / 64, HH, BB), 256, 0, stream>>>(qkv, w1u, w2u, concat);

    // 4) depthwise update + conv -> concat[:, 768:832]
    ttt_dwc_kernel<<<dim3(BB * DD), 256, 0, stream>>>(qkv, w3, concat);

    // 5) out = concat @ proj_W + proj_b   M=32768 N=768 K=832
    gemm_bias_f32<<<dim3(CC / 64, (BB * NN) / 128), 256, 0, stream>>>(
        concat, proj_W, proj_b, out, BB * NN, CC, CONC);
}